// ExpEig_16595753632482
// MI455X (gfx1250) — compile-verified
//
#include <hip/hip_runtime.h>
#include <hip/hip_bf16.h>

typedef float v2f __attribute__((ext_vector_type(2)));
typedef float v8f __attribute__((ext_vector_type(8)));

#define WAVES_PER_BLOCK 8
#define MAT_ELEMS 1024   // 32*32

// C = A(32x32) * B(32x32), both row-major in LDS. Result in 4 v8f accumulator
// tiles (tile t = i*2+j covers rows 16i..16i+15, cols 16j..16j+15) laid out in
// the CDNA5 WMMA 16x16 f32 C/D VGPR format.
__device__ __forceinline__ void mm32x32(const float* a, const float* b,
                                        v8f acc[4], int lane)
{
    const int m  = lane & 15;          // A row / B col within tile
    const int kh = (lane >> 4) << 1;   // 0 or 2: K sub-pair per half-wave
    #pragma unroll
    for (int i = 0; i < 2; ++i) {
        // A fragments for this 16-row block: 8 k-steps x (16x4 fp32 frag)
        v2f af[8];
        #pragma unroll
        for (int kk = 0; kk < 8; ++kk) {
            // A layout (ISA 7.12.2, 32-bit A 16x4): lane m holds (m, kh) and
            // (m, kh+1) -> contiguous pair in row-major storage.
            af[kk] = *(const v2f*)(a + (16*i + m)*32 + 4*kk + kh);
        }
        #pragma unroll
        for (int j = 0; j < 2; ++j) {
            v8f c = {};
            #pragma unroll
            for (int kk = 0; kk < 8; ++kk) {
                const int kb = 4*kk + kh;
                // B layout (4x16): VGPR0 = rows {0,2}, VGPR1 = rows {1,3}
                v2f bf;
                bf.x = b[(kb    )*32 + 16*j + m];
                bf.y = b[(kb + 1)*32 + 16*j + m];
                c = __builtin_amdgcn_wmma_f32_16x16x4_f32(
                        /*neg_a=*/false, af[kk], /*neg_b=*/false, bf,
                        /*c_mod=*/(short)0, c, /*reuse_a=*/false, /*reuse_b=*/false);
            }
            acc[i*2 + j] = c;
        }
    }
}

// Write accumulator tiles back to a row-major 32x32 LDS buffer:
// dst = acc * mul (+ I if addI)
__device__ __forceinline__ void store_tiles_lds(float* dst, const v8f acc[4],
                                                int lane, float mul, bool addI)
{
    const int n  = lane & 15;
    const int hi = (lane >> 4) * 8;    // C/D layout: lanes 16-31 hold M = r+8
    #pragma unroll
    for (int i = 0; i < 2; ++i)
        #pragma unroll
        for (int j = 0; j < 2; ++j)
            #pragma unroll
            for (int r = 0; r < 8; ++r) {
                const int row = 16*i + hi + r;
                const int col = 16*j + n;
                float v = acc[i*2 + j][r] * mul;
                if (addI && row == col) v += 1.0f;
                dst[row*32 + col] = v;
            }
}

// exp(P) via scaling-and-squaring: X = P/32; T = order-8 Taylor of exp(X)
// by Horner; then square 5 times. One wave32 per matrix.
__global__ __launch_bounds__(32 * WAVES_PER_BLOCK)
void expeig_kernel(const float* __restrict__ P, float* __restrict__ out)
{
    __shared__ float smem[WAVES_PER_BLOCK * 2 * MAT_ELEMS];  // 64 KB

    const int lane = threadIdx.x & 31;
    const int wave = threadIdx.x >> 5;
    const int gm   = blockIdx.x * WAVES_PER_BLOCK + wave;

    float* Xs = smem + wave * 2 * MAT_ELEMS;   // scaled input X
    float* Ts = Xs + MAT_ELEMS;                // Horner accumulator T

    const float* Pm = P + (size_t)gm * MAT_ELEMS;
    const float scale = 1.0f / 32.0f;          // 2^-5

    // Load P (coalesced b128), scale -> X; init T = I + X/8 (Horner seed).
    #pragma unroll
    for (int v = 0; v < 8; ++v) {
        const int e = v * 128 + lane * 4;      // flat element index, 16B aligned
        float4 p = *(const float4*)(Pm + e);
        float4 x = make_float4(p.x*scale, p.y*scale, p.z*scale, p.w*scale);
        float4 t = make_float4(x.x*0.125f, x.y*0.125f, x.z*0.125f, x.w*0.125f);
        const int row = e >> 5, col = e & 31;
        const int d = row - col;               // diagonal lands in [0,4)
        if      (d == 0) t.x += 1.0f;
        else if (d == 1) t.y += 1.0f;
        else if (d == 2) t.z += 1.0f;
        else if (d == 3) t.w += 1.0f;
        *(float4*)(Xs + e) = x;
        *(float4*)(Ts + e) = t;
    }

    v8f acc[4];

    // Horner: for k = 7..1:  T = I + (X*T)/k
    #pragma unroll
    for (int k = 7; k >= 1; --k) {
        mm32x32(Xs, Ts, acc, lane);
        store_tiles_lds(Ts, acc, lane, 1.0f / (float)k, true);
    }

    // 5 squarings: T = T*T  (undo the 2^-5 scaling)
    #pragma unroll
    for (int sq = 0; sq < 5; ++sq) {
        mm32x32(Ts, Ts, acc, lane);
        store_tiles_lds(Ts, acc, lane, 1.0f, false);
    }

    // Coalesced 128-bit write-out via LDS staging.
    float* Om = out + (size_t)gm * MAT_ELEMS;
    #pragma unroll
    for (int v = 0; v < 8; ++v) {
        const int e = v * 128 + lane * 4;
        *(float4*)(Om + e) = *(const float4*)(Ts + e);
    }
}

extern "C" void kernel_launch(void* const* d_in, const int* in_sizes, int n_in,
                              void* d_out, int out_size, void* d_ws, size_t ws_size,
                              hipStream_t stream)
{
    const float* P = (const float*)d_in[0];
    float* out = (float*)d_out;
    const int nmat   = in_sizes[0] / MAT_ELEMS;        // 2048*16 = 32768
    const int blocks = nmat / WAVES_PER_BLOCK;         // 4096
    expeig_kernel<<<blocks, 32 * WAVES_PER_BLOCK, 0, stream>>>(P, out);
}